// MoEModel_72361609003370
// MI455X (gfx1250) — compile-verified
//
#include <hip/hip_runtime.h>

typedef float v2f __attribute__((ext_vector_type(2)));
typedef float v8f __attribute__((ext_vector_type(8)));

#define BB 65536
#define DD 784
#define EE 16
#define HH 64
#define OO 10
#define KC 16      // K-chunk staged in LDS (784 = 49 * 16)
#define NCH (DD / KC)
#define XS 20      // padded x-tile stride (floats): 16B-aligned cols, conflict-free A-frag reads
#define W1S 72     // padded W1-tile stride: 16B-aligned, lanes 0-15 vs 16-31 hit disjoint banks
#define HS 68      // padded h stride: 4m mod 64 distinct for m=0..15

// ---------------------------------------------------------------- zero counters
__global__ void moe_zero_cnt(unsigned int* cnt) {
  if (threadIdx.x < EE) cnt[threadIdx.x] = 0u;
}

// ---------------------------------------------------------------- gate + routing
// Each wave: one 16-token x 16-expert WMMA tile. Block = 4 waves = 64 tokens.
// Double-buffered per-wave x staging => no barriers in the K loop.
__global__ __launch_bounds__(128) void moe_gate(
    const float* __restrict__ x, const float* __restrict__ gW,
    const float* __restrict__ gb,
    unsigned int* __restrict__ cnt,
    unsigned int* __restrict__ btok,   // [E][B] packed: (token<<1)|slot
    float* __restrict__ bwgt)          // [E][B]
{
  __shared__ float sGW[DD * EE];           // 50176 B
  __shared__ float sX[4][2 * 16 * XS];     // per-wave double-buffered x chunk (logits alias after loop)

  const int tid  = threadIdx.x;
  const int lane = tid & 31;
  const int wv   = tid >> 5;
  const int m    = lane & 15;              // token row / expert col within tile
  const int kp   = lane >> 4;              // K-pair select (0 or 1)

  for (int i = tid; i < DD * EE; i += 128) sGW[i] = gW[i];

  const int tok0 = blockIdx.x * 64 + wv * 16;
  const int rx = lane >> 1, cx = (lane & 1) * 8;     // staging: 2 lanes/row, 8 floats each
  const float* xrow = x + (size_t)(tok0 + rx) * DD + cx;

  float* sb0 = sX[wv];
  float* sb1 = sX[wv] + 16 * XS;

  // preload chunk 0 (per-wave buffer: same-wave LDS ordering suffices)
  {
    const float4* gp = (const float4*)(xrow);
    float4 a0 = gp[0], a1 = gp[1];
    *(float4*)(sb0 + rx * XS + cx)     = a0;
    *(float4*)(sb0 + rx * XS + cx + 4) = a1;
  }
  __syncthreads();   // sGW visibility

  v8f acc = {};
  for (int c = 0; c < NCH; ++c) {
    const int kc = c * KC;
    float4 p0 = {}, p1 = {};
    const bool hn = (c + 1 < NCH);
    if (hn) {
      const float4* gp = (const float4*)(xrow + kc + KC);
      p0 = gp[0]; p1 = gp[1];
      if (c + 2 < NCH) __builtin_prefetch(xrow + kc + 2 * KC, 0, 1);
    }
    const float* sx = (c & 1) ? sb1 : sb0;
#pragma unroll
    for (int kk = 0; kk < KC; kk += 4) {
      const int ka = kk + 2 * kp;          // chunk-local K index for this lane
      v2f a, b;
      a.x = sx[m * XS + ka];
      a.y = sx[m * XS + ka + 1];
      b.x = sGW[(kc + ka) * EE + m];       // B[k][n], n = lane&15
      b.y = sGW[(kc + ka + 1) * EE + m];
      acc = __builtin_amdgcn_wmma_f32_16x16x4_f32(false, a, false, b,
                                                  (short)0, acc, false, false);
    }
    if (hn) {
      float* sn = ((c + 1) & 1) ? sb1 : sb0;
      *(float4*)(sn + rx * XS + cx)     = p0;
      *(float4*)(sn + rx * XS + cx + 4) = p1;
    }
  }

  // add gate bias (column n = m) and dump logits[token][expert] into (dead) staging LDS
  float* sl = sX[wv];
  {
    const float bias = gb[m];
#pragma unroll
    for (int v = 0; v < 8; ++v) {
      const int row = v + 8 * kp;          // token row in tile
      sl[row * 16 + m] = acc[v] + bias;
    }
  }
  __syncthreads();

  // lanes 0..15: per-token top-2 + softmax(2), bucket by expert
  if (lane < 16) {
    const float* lg = sl + lane * 16;
    float v1 = -1e30f, v2 = -1e30f; int i1 = 0, i2 = 0;
#pragma unroll
    for (int e = 0; e < EE; ++e) {
      const float v = lg[e];
      if (v > v1)      { v2 = v1; i2 = i1; v1 = v; i1 = e; }
      else if (v > v2) { v2 = v;  i2 = e; }
    }
    const float ez = __expf(v2 - v1);
    const float w1 = 1.0f / (1.0f + ez);
    const float w2 = ez * w1;
    const unsigned t = (unsigned)(tok0 + lane);
    unsigned s1 = atomicAdd(&cnt[i1], 1u);
    btok[(size_t)i1 * BB + s1] = (t << 1) | 0u;
    bwgt[(size_t)i1 * BB + s1] = w1;
    unsigned s2 = atomicAdd(&cnt[i2], 1u);
    btok[(size_t)i2 * BB + s2] = (t << 1) | 1u;
    bwgt[(size_t)i2 * BB + s2] = w2;
  }
}

// ---------------------------------------------------------------- routed expert MLP
// grid = (1024 chunks, 16 experts); block = 4 waves x 16 gathered tokens.
// Double-buffered LDS (W1 block-shared + per-wave gathered x): 1 barrier / chunk.
__global__ __launch_bounds__(128) void moe_expert(
    const float* __restrict__ x,
    const float* __restrict__ W1, const float* __restrict__ b1,
    const float* __restrict__ W2, const float* __restrict__ b2,
    const unsigned int* __restrict__ cnt,
    const unsigned int* __restrict__ btok,
    const float* __restrict__ bwgt,
    float* __restrict__ yp)             // [B][2][O] partials
{
  const int e = blockIdx.y;
  const unsigned n_e = cnt[e];
  const unsigned base = blockIdx.x * 64;
  if (base >= n_e) return;

  __shared__ float sW1[2 * KC * W1S];      // block-shared W1 K-chunk, double buffered
  __shared__ float sXg[4][2 * 16 * XS];    // per-wave gathered x chunk, double buffered
  __shared__ float sH[4][16 * HS];         // per-wave hidden activations (16 x 64)
  __shared__ unsigned sTok[4][16];
  __shared__ float sWgt[4][16];

  const int tid  = threadIdx.x;
  const int lane = tid & 31;
  const int wv   = tid >> 5;
  const int m    = lane & 15;
  const int kp   = lane >> 4;

  if (lane < 16) {
    const unsigned s = base + wv * 16 + lane;
    unsigned pk = 0u; float w = 0.f;
    if (s < n_e) { pk = btok[(size_t)e * BB + s]; w = bwgt[(size_t)e * BB + s]; }
    sTok[wv][lane] = pk;
    sWgt[wv][lane] = w;
  }
  __syncthreads();

  const float* W1e = W1 + (size_t)e * DD * HH;
  const int rw = tid >> 3,  cw = (tid & 7) * 8;    // W1 staging: 8 threads/row
  const int rx = lane >> 1, cx = (lane & 1) * 8;   // x staging: 2 lanes/row
  const unsigned trow = sTok[wv][rx] >> 1;
  const float* xrow = x + (size_t)trow * DD + cx;

  float* swb0 = sW1;       float* swb1 = sW1 + KC * W1S;
  float* sxb0 = sXg[wv];   float* sxb1 = sXg[wv] + 16 * XS;

  // preload chunk 0
  {
    const float4* gw = (const float4*)(W1e + (size_t)rw * HH + cw);
    float4 w0 = gw[0], w1v = gw[1];
    *(float4*)(&swb0[rw * W1S + cw])     = w0;
    *(float4*)(&swb0[rw * W1S + cw + 4]) = w1v;
    const float4* gx = (const float4*)(xrow);
    float4 x0 = gx[0], x1 = gx[1];
    *(float4*)(sxb0 + rx * XS + cx)     = x0;
    *(float4*)(sxb0 + rx * XS + cx + 4) = x1;
  }
  __syncthreads();

  v8f acc[4] = {{}, {}, {}, {}};
  for (int c = 0; c < NCH; ++c) {
    const int kc = c * KC;
    float4 wA = {}, wB = {}, xA = {}, xB = {};
    const bool hn = (c + 1 < NCH);
    if (hn) {
      const float4* gw = (const float4*)(W1e + (size_t)(kc + KC + rw) * HH + cw);
      wA = gw[0]; wB = gw[1];
      const float4* gx = (const float4*)(xrow + kc + KC);
      xA = gx[0]; xB = gx[1];
      if (c + 2 < NCH) {
        __builtin_prefetch(W1e + (size_t)(kc + 2 * KC + rw) * HH + cw, 0, 1);
        __builtin_prefetch(xrow + kc + 2 * KC, 0, 1);
      }
    }
    const float* sw = (c & 1) ? swb1 : swb0;
    const float* sx = (c & 1) ? sxb1 : sxb0;
#pragma unroll
    for (int kk = 0; kk < KC; kk += 4) {
      const int ka = kk + 2 * kp;
      v2f a;
      a.x = sx[m * XS + ka];
      a.y = sx[m * XS + ka + 1];
#pragma unroll
      for (int nt = 0; nt < 4; ++nt) {
        v2f b;
        b.x = sw[ka * W1S + nt * 16 + m];
        b.y = sw[(ka + 1) * W1S + nt * 16 + m];
        acc[nt] = __builtin_amdgcn_wmma_f32_16x16x4_f32(false, a, false, b,
                                                        (short)0, acc[nt],
                                                        false, false);
      }
    }
    if (hn) {
      float* swn = ((c + 1) & 1) ? swb1 : swb0;
      float* sxn = ((c + 1) & 1) ? sxb1 : sxb0;
      *(float4*)(&swn[rw * W1S + cw])     = wA;
      *(float4*)(&swn[rw * W1S + cw + 4]) = wB;
      *(float4*)(sxn + rx * XS + cx)     = xA;
      *(float4*)(sxn + rx * XS + cx + 4) = xB;
    }
    __syncthreads();
  }

  // bias + ReLU, transpose h into LDS for the second GEMM's A-fragments
  {
    const float* b1e = b1 + e * HH;
    float* sh = sH[wv];
#pragma unroll
    for (int nt = 0; nt < 4; ++nt) {
      const float bv = b1e[nt * 16 + m];
#pragma unroll
      for (int v = 0; v < 8; ++v) {
        const int row = v + 8 * kp;
        const float h = acc[nt][v] + bv;
        sh[row * HS + nt * 16 + m] = h > 0.f ? h : 0.f;
      }
    }
  }
  __syncthreads();

  // layer 2: y[16x10] = h[16x64] @ W2[e] (N padded to 16 with zero columns)
  const float* W2e = W2 + e * HH * OO;
  const bool nvalid = (m < OO);
  const float* sh = sH[wv];
  v8f oacc = {};
#pragma unroll
  for (int ks = 0; ks < HH; ks += 4) {
    const int ka = ks + 2 * kp;
    v2f a, b;
    a.x = sh[m * HS + ka];
    a.y = sh[m * HS + ka + 1];
    b.x = nvalid ? W2e[ka * OO + m] : 0.f;
    b.y = nvalid ? W2e[(ka + 1) * OO + m] : 0.f;
    oacc = __builtin_amdgcn_wmma_f32_16x16x4_f32(false, a, false, b,
                                                 (short)0, oacc, false, false);
  }

  // weighted partials to fixed [token][slot] locations (order-independent => deterministic)
  if (nvalid) {
    const float b2v = b2[e * OO + m];
#pragma unroll
    for (int v = 0; v < 8; ++v) {
      const int row = v + 8 * kp;
      const unsigned s = base + wv * 16 + (unsigned)row;
      if (s < n_e) {
        const unsigned pk = sTok[wv][row];
        const unsigned t = pk >> 1, slot = pk & 1u;
        yp[((size_t)t * 2 + slot) * OO + m] = sWgt[wv][row] * (oacc[v] + b2v);
      }
    }
  }
}

// ---------------------------------------------------------------- combine 2 partials
__global__ void moe_combine(const float* __restrict__ yp,
                            float* __restrict__ out, int n) {
  const int i = blockIdx.x * blockDim.x + threadIdx.x;
  if (i < n) {
    const int b = i / OO, o = i % OO;
    out[i] = yp[((size_t)b * 2) * OO + o] + yp[((size_t)b * 2 + 1) * OO + o];
  }
}

// ---------------------------------------------------------------- launch
extern "C" void kernel_launch(void* const* d_in, const int* in_sizes, int n_in,
                              void* d_out, int out_size, void* d_ws, size_t ws_size,
                              hipStream_t stream) {
  (void)in_sizes; (void)n_in; (void)out_size; (void)ws_size;
  const float* x  = (const float*)d_in[0];
  const float* gW = (const float*)d_in[1];
  const float* gb = (const float*)d_in[2];
  const float* W1 = (const float*)d_in[3];
  const float* b1 = (const float*)d_in[4];
  const float* W2 = (const float*)d_in[5];
  const float* b2 = (const float*)d_in[6];
  float* out = (float*)d_out;

  char* ws = (char*)d_ws;
  unsigned int* cnt  = (unsigned int*)ws;                                   // 16 u32 (pad 256B)
  unsigned int* btok = (unsigned int*)(ws + 256);                           // 4 MB
  float*        bwgt = (float*)(ws + 256 + (size_t)EE * BB * 4);            // 4 MB
  float*        yp   = (float*)(ws + 256 + (size_t)2 * EE * BB * 4);        // 5.25 MB

  moe_zero_cnt<<<1, 32, 0, stream>>>(cnt);
  moe_gate<<<BB / 64, 128, 0, stream>>>(x, gW, gb, cnt, btok, bwgt);
  moe_expert<<<dim3(BB / 64, EE), 128, 0, stream>>>(x, W1, b1, W2, b2,
                                                    cnt, btok, bwgt, yp);
  moe_combine<<<(BB * OO + 255) / 256, 256, 0, stream>>>(yp, out, BB * OO);
}